// MoEOutputLayer_58351425684015
// MI455X (gfx1250) — compile-verified
//
#include <hip/hip_runtime.h>

// einsum('bsn,bnse->bse'): B=8, S=4096, N=8, E=128, all f32.
// Bandwidth-bound (152 MB, ~6.5us floor @ 23.3 TB/s). One wave per (b,s):
// the N=8 weighted combine is done exactly with chained V_WMMA_F32_16X16X4_F32
// (A = w-row replicated over all 16 M rows -> every D row equals the result).

typedef __attribute__((ext_vector_type(2))) float v2f;
typedef __attribute__((ext_vector_type(8))) float v8f;

#define BB 8
#define SS 4096
#define NN 8
#define EE 128

__global__ __launch_bounds__(256) void moe_combine_wmma(
    const float* __restrict__ W,   // [B,S,N]   routing_logits
    const float* __restrict__ X,   // [B,N,S,E] expert_outputs
    float* __restrict__ O)         // [B,S,E]
{
    const int lane = threadIdx.x & 31;
    const int wave = threadIdx.x >> 5;
    const int g    = blockIdx.x * 8 + wave;   // flat (b*S + s), 0..32767
    const int b    = g >> 12;                 // / 4096
    const int s    = g & (SS - 1);

    const int half = lane >> 4;               // 0: lanes 0-15, 1: lanes 16-31
    const int col  = lane & 15;

    // ---- A operands: weight row w[s,0..7], replicated across all 16 M rows.
    // A 16x4 f32 layout: VGPR0 = K0 (lanes 0-15) / K2 (lanes 16-31), VGPR1 = K1/K3.
    const float* wrow = W + (size_t)g * NN;
    const float4 wa = ((const float4*)wrow)[0];   // w0..w3 (broadcast read, 16B aligned)
    const float4 wb = ((const float4*)wrow)[1];   // w4..w7
    v2f A0, A1;
    A0.x = half ? wa.z : wa.x;    // K = 2 : 0
    A0.y = half ? wa.w : wa.y;    // K = 3 : 1
    A1.x = half ? wb.z : wb.x;    // K = 6 : 4
    A1.y = half ? wb.w : wb.y;    // K = 7 : 5

    // ---- B operands: x[k, s, e0+col] in native 4x16 layout.
    // VGPR0 holds K=0 (lanes 0-15) and K=2 (lanes 16-31); VGPR1 holds K=1/K=3.
    const size_t nstride = (size_t)SS * EE;                      // stride between n planes
    const float* xbase = X + (size_t)b * NN * nstride + (size_t)s * EE + col;
    const float* p0 = xbase + (size_t)(2 * half + 0) * nstride;  // K = 0 / 2
    const float* p1 = xbase + (size_t)(2 * half + 1) * nstride;  // K = 1 / 3
    const float* p4 = xbase + (size_t)(2 * half + 4) * nstride;  // K = 4 / 6
    const float* p5 = xbase + (size_t)(2 * half + 5) * nstride;  // K = 5 / 7

    float* orow = O + (size_t)g * EE;

#pragma unroll
    for (int tp = 0; tp < 4; ++tp) {           // pairs of 16-wide e-tiles
        const int e0 = tp * 32;

        // e-tile [e0, e0+16): D[m, c] = sum_{n=0..7} w[n] * x[n, s, e0+c]  (all m equal)
        v2f B0, B1;
        v8f c0 = {};
        B0.x = p0[e0];      B0.y = p1[e0];
        B1.x = p4[e0];      B1.y = p5[e0];
        c0 = __builtin_amdgcn_wmma_f32_16x16x4_f32(false, A0, false, B0, (short)0, c0, false, false);
        c0 = __builtin_amdgcn_wmma_f32_16x16x4_f32(false, A1, false, B1, (short)0, c0, false, false);

        // e-tile [e0+16, e0+32)
        v8f c1 = {};
        B0.x = p0[e0 + 16]; B0.y = p1[e0 + 16];
        B1.x = p4[e0 + 16]; B1.y = p5[e0 + 16];
        c1 = __builtin_amdgcn_wmma_f32_16x16x4_f32(false, A0, false, B0, (short)0, c1, false, false);
        c1 = __builtin_amdgcn_wmma_f32_16x16x4_f32(false, A1, false, B1, (short)0, c1, false, false);

        // D VGPR0: lanes 0-15 = row M=0 (N=col), lanes 16-31 = row M=8 (N=col);
        // all rows identical, so a full-wave 128B coalesced store covers both tiles.
        const float val = half ? c1[0] : c0[0];
        __builtin_nontemporal_store(val, orow + e0 + lane);
    }
}

extern "C" void kernel_launch(void* const* d_in, const int* in_sizes, int n_in,
                              void* d_out, int out_size, void* d_ws, size_t ws_size,
                              hipStream_t stream) {
    const float* W = (const float*)d_in[0];   // routing_logits [B,S,N]
    const float* X = (const float*)d_in[1];   // expert_outputs [B,N,S,E]
    float* O = (float*)d_out;                 // [B,S,E]
    (void)in_sizes; (void)n_in; (void)out_size; (void)d_ws; (void)ws_size;

    dim3 grid(BB * SS / 8);   // 4096 blocks x 8 waves = one wave per (b,s)
    dim3 block(256);
    hipLaunchKernelGGL(moe_combine_wmma, grid, block, 0, stream, W, X, O);
}